// newcell_48455821033529
// MI455X (gfx1250) — compile-verified
//
#include <hip/hip_runtime.h>
#include <hip/hip_bf16.h>

// Problem dims (from reference): B=64, C=64, H=W=56, MID=128
#define NB   64
#define NC   64
#define NH   56
#define NW   56
#define HW   (NH * NW)      // 3136
#define NMID 128
#define EPS  1e-5f

typedef __attribute__((ext_vector_type(2))) float v2f;
typedef __attribute__((ext_vector_type(8))) float v8f;

// ---------------------------------------------------------------------------
// Fused 1x1-conv + BatchNorm(eval) + ReLU as a WMMA f32 GEMM.
//   out[b, o, p] = relu( (sum_c w[o,c]*in[b,c,p]) * sc[o] + (beta[o]-mean[o]*sc[o]) )
// BN scale sc[o] is folded into the weights when staging them in LDS.
// M = 64 output channels, K = KDIM input channels, N = HW pixels.
// Block = 128 threads (4 waves). Each wave computes a 64(M) x 16(N) strip:
// 4 accumulators (v8f each), 4 x V_WMMA_F32_16X16X4_F32 per K-step of 4.
// Grid = (HW/64, NB).
// ---------------------------------------------------------------------------
template <int KDIM>
__global__ __launch_bounds__(128) void conv1x1_bn_relu(
    const float* __restrict__ in,   // [NB, KDIM, HW]
    const float* __restrict__ w,    // [64, KDIM] row-major (o, c)
    const float* __restrict__ g,    // gamma [64]
    const float* __restrict__ bta,  // beta  [64]
    const float* __restrict__ mn,   // mean  [64]
    const float* __restrict__ vr,   // var   [64]
    float* __restrict__ out)        // [NB, 64, HW]
{
    __shared__ float lds_w[64 * KDIM];
    __shared__ float lds_bias[64];

    const int tid = threadIdx.x;

    // Stage BN-folded weights into LDS.
    for (int i = tid; i < 64 * KDIM; i += 128) {
        const int o = i / KDIM;
        const float sc = g[o] * rsqrtf(vr[o] + EPS);
        lds_w[i] = w[i] * sc;
    }
    if (tid < 64) {
        const float sc = g[tid] * rsqrtf(vr[tid] + EPS);
        lds_bias[tid] = bta[tid] - mn[tid] * sc;
    }
    __syncthreads();

    const int wave = tid >> 5;
    const int lane = tid & 31;
    const int hi   = lane >> 4;   // 0: lanes 0-15, 1: lanes 16-31
    const int lo   = lane & 15;

    const int b  = blockIdx.y;
    const int p0 = (blockIdx.x * 4 + wave) * 16;   // pixel tile base (HW % 64 == 0)

    const float* inB = in + (size_t)b * KDIM * HW + p0 + lo;

    v8f acc0 = {}, acc1 = {}, acc2 = {}, acc3 = {};

    for (int kk = 0; kk < KDIM; kk += 4) {
        // B tile (4x16, K x N): vgpr j holds K = kk + j + 2*hi, N = lo.
        const int c0 = kk + 2 * hi;
        v2f bm;
        bm.x = inB[(size_t)c0 * HW];
        bm.y = inB[(size_t)(c0 + 1) * HW];

        // A tiles (16x4, M x K): lane lo is row M, vgpr j holds K = kk + j + 2*hi.
        const int krow = kk + 2 * hi;   // even -> 8B-aligned LDS float2 loads
        const v2f a0 = *(const v2f*)&lds_w[(0 * 16 + lo) * KDIM + krow];
        const v2f a1 = *(const v2f*)&lds_w[(1 * 16 + lo) * KDIM + krow];
        const v2f a2 = *(const v2f*)&lds_w[(2 * 16 + lo) * KDIM + krow];
        const v2f a3 = *(const v2f*)&lds_w[(3 * 16 + lo) * KDIM + krow];

        acc0 = __builtin_amdgcn_wmma_f32_16x16x4_f32(false, a0, false, bm, (short)0, acc0, false, false);
        acc1 = __builtin_amdgcn_wmma_f32_16x16x4_f32(false, a1, false, bm, (short)0, acc1, false, false);
        acc2 = __builtin_amdgcn_wmma_f32_16x16x4_f32(false, a2, false, bm, (short)0, acc2, false, false);
        acc3 = __builtin_amdgcn_wmma_f32_16x16x4_f32(false, a3, false, bm, (short)0, acc3, false, false);
    }

    // Epilogue: D layout is M = vgpr + 8*hi, N = lo. Stores coalesce per row.
    float* outB = out + (size_t)b * 64 * HW + p0 + lo;
#pragma unroll
    for (int v = 0; v < 8; ++v) {
        const int o0 = 0 * 16 + v + 8 * hi;
        const int o1 = 1 * 16 + v + 8 * hi;
        const int o2 = 2 * 16 + v + 8 * hi;
        const int o3 = 3 * 16 + v + 8 * hi;
        outB[(size_t)o0 * HW] = fmaxf(acc0[v] + lds_bias[o0], 0.0f);
        outB[(size_t)o1 * HW] = fmaxf(acc1[v] + lds_bias[o1], 0.0f);
        outB[(size_t)o2 * HW] = fmaxf(acc2[v] + lds_bias[o2], 0.0f);
        outB[(size_t)o3 * HW] = fmaxf(acc3[v] + lds_bias[o3], 0.0f);
    }
}

// ---------------------------------------------------------------------------
// Pool stage. One block per (b, c) plane of node0.
// Phase 1: node1 = maxpool3(node0) and node2 = avgpool3(node0) into LDS
//          (node2 also written to cat channel c).
// Phase 2: node3 = maxpool3(node1) + maxpool3(node2) -> cat channel 64+c.
// Out-of-plane is -inf for max (matches reduce_window -inf padding) and
// contributes 0 to the sum with a fixed /9 (count_include_pad).
// ---------------------------------------------------------------------------
__global__ __launch_bounds__(256) void pool_stage(
    const float* __restrict__ node0,   // [NB, 64, HW]
    float* __restrict__ cat)           // [NB, 128, HW]
{
    __shared__ float s_max[HW];
    __shared__ float s_avg[HW];

    const int bc = blockIdx.x;            // b*64 + c
    const int b  = bc >> 6;
    const int c  = bc & 63;

    const float* plane = node0 + (size_t)bc * HW;
    float* out2 = cat + ((size_t)b * NMID + c) * HW;        // node2 slot
    float* out3 = cat + ((size_t)b * NMID + 64 + c) * HW;   // node3 slot

    for (int i = threadIdx.x; i < HW; i += 256) {
        const int y = i / NW, x = i - y * NW;
        float mx = -INFINITY;
        float sm = 0.0f;
#pragma unroll
        for (int dy = -1; dy <= 1; ++dy) {
#pragma unroll
            for (int dx = -1; dx <= 1; ++dx) {
                const int yy = y + dy, xx = x + dx;
                if (yy >= 0 && yy < NH && xx >= 0 && xx < NW) {
                    const float val = plane[yy * NW + xx];
                    mx = fmaxf(mx, val);
                    sm += val;
                }
            }
        }
        const float avg = sm * (1.0f / 9.0f);
        s_max[i] = mx;
        s_avg[i] = avg;
        out2[i] = avg;
    }
    __syncthreads();

    for (int i = threadIdx.x; i < HW; i += 256) {
        const int y = i / NW, x = i - y * NW;
        float m1 = -INFINITY;   // maxpool3(node1)
        float m2 = -INFINITY;   // maxpool3(node2)
#pragma unroll
        for (int dy = -1; dy <= 1; ++dy) {
#pragma unroll
            for (int dx = -1; dx <= 1; ++dx) {
                const int yy = y + dy, xx = x + dx;
                if (yy >= 0 && yy < NH && xx >= 0 && xx < NW) {
                    const int j = yy * NW + xx;
                    m1 = fmaxf(m1, s_max[j]);
                    m2 = fmaxf(m2, s_avg[j]);
                }
            }
        }
        out3[i] = m1 + m2;
    }
}

// ---------------------------------------------------------------------------
// Launch: GEMM1 (x -> node0 in ws), pools (node0 -> cat in ws), GEMM2
// (cat -> out). node0 (51 MB) and cat (103 MB) stay L2-resident (192 MB L2),
// so HBM traffic is ~x + out = 103 MB -> ~4.4 us floor at 23.3 TB/s.
// ---------------------------------------------------------------------------
extern "C" void kernel_launch(void* const* d_in, const int* in_sizes, int n_in,
                              void* d_out, int out_size, void* d_ws, size_t ws_size,
                              hipStream_t stream) {
    const float* x     = (const float*)d_in[0];   // [64,64,56,56]
    const float* w1    = (const float*)d_in[1];   // [64,64]  (o, c)
    const float* w2    = (const float*)d_in[2];   // [64,128] (o, c)
    const float* bn1_g = (const float*)d_in[3];
    const float* bn1_b = (const float*)d_in[4];
    const float* bn1_m = (const float*)d_in[5];
    const float* bn1_v = (const float*)d_in[6];
    const float* bn2_g = (const float*)d_in[7];
    const float* bn2_b = (const float*)d_in[8];
    const float* bn2_m = (const float*)d_in[9];
    const float* bn2_v = (const float*)d_in[10];
    float* out = (float*)d_out;

    float* node0 = (float*)d_ws;                               // 64*64*3136 f32 (51 MB)
    float* cat   = node0 + (size_t)NB * NC * HW;               // 64*128*3136 f32 (103 MB)

    const dim3 gemm_grid(HW / 64, NB);   // 49 x 64
    const dim3 gemm_block(128);

    conv1x1_bn_relu<NC><<<gemm_grid, gemm_block, 0, stream>>>(
        x, w1, bn1_g, bn1_b, bn1_m, bn1_v, node0);

    pool_stage<<<dim3(NB * NC), dim3(256), 0, stream>>>(node0, cat);

    conv1x1_bn_relu<NMID><<<gemm_grid, gemm_block, 0, stream>>>(
        cat, w2, bn2_g, bn2_b, bn2_m, bn2_v, out);
}